// RKNCell_10926396801277
// MI455X (gfx1250) — compile-verified
//
#include <hip/hip_runtime.h>
#include <math.h>

#define LOD   64
#define LSD   128
#define NBASIS 15
#define BATCH 4096
#define HID   64
#define NE    436
#define BWD   3
#define RPB   16      // batch rows per block
#define THREADS 128
#define CHUNK_I 16    // output rows per band chunk
#define CEMAX 112     // max band elements per 16-row chunk

typedef __attribute__((ext_vector_type(2))) float v2f;
typedef __attribute__((ext_vector_type(8))) float v8f;

#if defined(__has_builtin)
#if __has_builtin(__builtin_amdgcn_global_load_async_to_lds_b128)
#define HAVE_ASYNC_LDS 1
#endif
#endif

#ifdef HAVE_ASYNC_LDS
// exact pointee type from the hipcc diagnostic: int __attribute__((vector_size(16)))
typedef int v4i_vs __attribute__((vector_size(4 * sizeof(int))));
typedef __attribute__((address_space(1))) v4i_vs* gptr128;   // global  (AS1)
typedef __attribute__((address_space(3))) v4i_vs* lptr128;   // LDS     (AS3)
#endif

// output offsets (floats) in d_out, reference return order
#define OFF_PM   0
#define OFF_PCU  (BATCH*LSD)
#define OFF_PCL  (OFF_PCU + BATCH*LOD)
#define OFF_PCS  (OFF_PCL + BATCH*LOD)
#define OFF_NM   (OFF_PCS + BATCH*LOD)
#define OFF_NCU  (OFF_NM + BATCH*LSD)
#define OFF_NCL  (OFF_NCU + BATCH*LOD)
#define OFF_NCS  (OFF_NCL + BATCH*LOD)

// LDS layout (float offsets). Weights stored PAIR-INTERLEAVED [k/2][n][2] so a
// WMMA B-fragment (W^T[k][n], W^T[k+1][n]) is a single aligned ds_load_b64.
#define O_PM   0                    // 16*128
#define O_PCU  (O_PM + RPB*LSD)     // 16*64
#define O_PCL  (O_PCU + RPB*LOD)
#define O_PCS  (O_PCL + RPB*LOD)
#define O_W1   (O_PCS + RPB*LOD)    // 128*64
#define O_W2   (O_W1 + LSD*HID)     // 64*64
#define O_W3   (O_W2 + HID*HID)     // 64*16 (n padded to 16)
#define O_B1   (O_W3 + HID*16)      // 64
#define O_B2   (O_B1 + HID)         // 64
#define O_B3   (O_B2 + HID)         // 16
#define O_C    (O_B3 + 16)          // 16*16 softmax coeffs
#define O_H    (O_C + RPB*16)       // 16*64 activations
#define O_TM   (O_H + RPB*HID)      // 4*16*112 expanded band chunk
#define O_BAS  (O_TM + 4*RPB*CEMAX) // 4*15*436 basis (async-staged)
#define O_RS   (O_BAS + 4*NBASIS*NE)// 65 ints (row_start)
#define O_DIAG (O_RS + (LOD+1))     // 436 ints (diag flag per band elem)
#define LDS_FLOATS (O_DIAG + NE)

__global__ __launch_bounds__(THREADS)
void rkn_cell_fused(const float* __restrict__ prior_mean,
                    const float* __restrict__ cov_u,
                    const float* __restrict__ cov_l,
                    const float* __restrict__ cov_s,
                    const float* __restrict__ obs,
                    const float* __restrict__ obs_var,
                    const float* __restrict__ W1, const float* __restrict__ b1,
                    const float* __restrict__ W2, const float* __restrict__ b2,
                    const float* __restrict__ W3, const float* __restrict__ b3,
                    const float* __restrict__ tm11b, const float* __restrict__ tm12b,
                    const float* __restrict__ tm21b, const float* __restrict__ tm22b,
                    const float* __restrict__ ltn,
                    float* __restrict__ out)
{
    extern __shared__ float smem[];
    float* sPM  = smem + O_PM;
    float* sPCU = smem + O_PCU;
    float* sPCL = smem + O_PCL;
    float* sPCS = smem + O_PCS;
    float* sW1  = smem + O_W1;
    float* sW2  = smem + O_W2;
    float* sW3  = smem + O_W3;
    float* sB1  = smem + O_B1;
    float* sB2  = smem + O_B2;
    float* sB3  = smem + O_B3;
    float* sC   = smem + O_C;
    float* sH   = smem + O_H;
    float* sTM  = smem + O_TM;
    float* sBAS = smem + O_BAS;
    int*   rs    = (int*)(smem + O_RS);
    int*   eDiag = (int*)(smem + O_DIAG);

    const int tid  = threadIdx.x;
    const int wave = tid >> 5;
    const int lane = tid & 31;
    const int r0   = blockIdx.x * RPB;

    // ---- async-stage ALL basis tensors into LDS; latency hidden behind the
    //      Kalman update + WMMA MLP (consumed only in phase 3/4) ----
#ifdef HAVE_ASYNC_LDS
    {
        const int VEC4 = (NBASIS * NE) / 4;  // 1635 b128 transfers per matrix
        #pragma unroll
        for (int m = 0; m < 4; ++m) {
            const float* src = (m == 0) ? tm11b : (m == 1) ? tm12b
                             : (m == 2) ? tm21b : tm22b;
            float* dst = sBAS + m * (NBASIS * NE);
            for (int q = tid; q < VEC4; q += THREADS) {
                __builtin_amdgcn_global_load_async_to_lds_b128(
                    (gptr128)(src + q * 4),
                    (lptr128)(dst + q * 4),
                    0, 0);
            }
        }
    }
#else
    for (int p = tid * 32; p < NBASIS * NE; p += THREADS * 32) {
        __builtin_prefetch(tm11b + p, 0, 1);
        __builtin_prefetch(tm12b + p, 0, 1);
        __builtin_prefetch(tm21b + p, 0, 1);
        __builtin_prefetch(tm22b + p, 0, 1);
    }
#endif

    // ---- phase 0: stage weights transposed + pair-interleaved, and tables ----
    for (int idx = tid; idx < HID * LSD; idx += THREADS) {
        int n = idx >> 7, k = idx & 127;
        sW1[(k >> 1) * (2 * HID) + n * 2 + (k & 1)] = W1[idx];
    }
    for (int idx = tid; idx < HID * HID; idx += THREADS) {
        int n = idx >> 6, k = idx & 63;
        sW2[(k >> 1) * (2 * HID) + n * 2 + (k & 1)] = W2[idx];
    }
    for (int idx = tid; idx < HID * 16; idx += THREADS) {
        int k = idx >> 4, n = idx & 15;
        sW3[(k >> 1) * 32 + n * 2 + (k & 1)] = (n < NBASIS) ? W3[n * HID + k] : 0.0f;
    }
    if (tid < HID) { sB1[tid] = b1[tid]; sB2[tid] = b2[tid]; }
    if (tid < NBASIS) sB3[tid] = b3[tid];
    for (int idx = tid; idx < NE; idx += THREADS) eDiag[idx] = 0;
    if (tid == 0) {
        int a = 0;
        for (int i = 0; i < LOD; ++i) {
            rs[i] = a;
            int jl = (i - BWD < 0) ? 0 : i - BWD;
            int jh = (i + BWD > LOD - 1) ? LOD - 1 : i + BWD;
            a += jh - jl + 1;
        }
        rs[LOD] = a; // == NE
    }

    // ---- phase 1: elementwise Kalman update ----
    for (int idx = tid; idx < RPB * LOD; idx += THREADS) {
        int r = idx >> 6, c = idx & 63;
        int row = r0 + r;
        float du = cov_u[row * LOD + c];
        float ov = obs_var[row * LOD + c];
        float sc = cov_s[row * LOD + c];
        float cl = cov_l[row * LOD + c];
        float pmu0 = prior_mean[row * LSD + c];
        float pml0 = prior_mean[row * LSD + LOD + c];
        float inv = 1.0f / (du + ov);
        float qu = du * inv;
        float ql = sc * inv;
        float res = obs[row * LOD + c] - pmu0;
        float pmu = pmu0 + qu * res;
        float pml = pml0 + ql * res;
        float cf = 1.0f - qu;
        float pcu = cf * du;
        float pcl = cl - ql * sc;
        float pcs = cf * sc;
        sPM[r * LSD + c] = pmu;
        sPM[r * LSD + LOD + c] = pml;
        sPCU[r * LOD + c] = pcu;
        sPCL[r * LOD + c] = pcl;
        sPCS[r * LOD + c] = pcs;
        out[OFF_PM + row * LSD + c] = pmu;
        out[OFF_PM + row * LSD + LOD + c] = pml;
        out[OFF_PCU + row * LOD + c] = pcu;
        out[OFF_PCL + row * LOD + c] = pcl;
        out[OFF_PCS + row * LOD + c] = pcs;
    }
    __syncthreads();
    if (tid < LOD) {
        int i = tid;
        int jl = (i - BWD < 0) ? 0 : i - BWD;
        eDiag[rs[i] + (i - jl)] = 1;
    }

    // ---- phase 2: MLP via V_WMMA_F32_16X16X4_F32 ----
    // 16x4 f32 A layout: lanes 0-15 hold M=lane with K=k0,k0+1 in v0,v1;
    //                    lanes 16-31 hold M=lane-16 with K=k0+2,k0+3.
    const int lm   = lane & 15;
    const int koff = (lane >> 4) << 1;   // 0 or 2
    const int mrow = (lane >> 4) << 3;   // D rows: 0..7 or 8..15

    { // layer 1: (16x128) @ (128x64), wave owns N-tile n0
        int n0 = wave << 4;
        v8f acc = {0.f,0.f,0.f,0.f,0.f,0.f,0.f,0.f};
        for (int ks = 0; ks < 32; ++ks) {
            int k = (ks << 2) + koff;          // even
            v2f a   = *(const v2f*)&sPM[lm * LSD + k];
            v2f bfr = *(const v2f*)&sW1[(k >> 1) * (2 * HID) + (n0 + lm) * 2];
            acc = __builtin_amdgcn_wmma_f32_16x16x4_f32(false, a, false, bfr,
                                                        (short)0, acc, false, false);
        }
        int n = n0 + lm;
        float bb = sB1[n];
        #pragma unroll
        for (int rr = 0; rr < 8; ++rr)
            sH[(mrow + rr) * HID + n] = tanhf(acc[rr] + bb);
    }
    __syncthreads();

    { // layer 2: (16x64) @ (64x64)
        int n0 = wave << 4;
        v8f acc = {0.f,0.f,0.f,0.f,0.f,0.f,0.f,0.f};
        for (int ks = 0; ks < 16; ++ks) {
            int k = (ks << 2) + koff;
            v2f a   = *(const v2f*)&sH[lm * HID + k];
            v2f bfr = *(const v2f*)&sW2[(k >> 1) * (2 * HID) + (n0 + lm) * 2];
            acc = __builtin_amdgcn_wmma_f32_16x16x4_f32(false, a, false, bfr,
                                                        (short)0, acc, false, false);
        }
        int n = n0 + lm;
        float bb = sB2[n];
        float t[8];
        #pragma unroll
        for (int rr = 0; rr < 8; ++rr) t[rr] = tanhf(acc[rr] + bb);
        __syncthreads();   // all 64 cols of sH consumed; safe to overwrite
        #pragma unroll
        for (int rr = 0; rr < 8; ++rr) sH[(mrow + rr) * HID + n] = t[rr];
    }
    __syncthreads();

    if (wave == 0) { // layer 3: (16x64) @ (64x15 padded to 16) -> logits
        v8f acc = {0.f,0.f,0.f,0.f,0.f,0.f,0.f,0.f};
        for (int ks = 0; ks < 16; ++ks) {
            int k = (ks << 2) + koff;
            v2f a   = *(const v2f*)&sH[lm * HID + k];
            v2f bfr = *(const v2f*)&sW3[(k >> 1) * 32 + lm * 2];
            acc = __builtin_amdgcn_wmma_f32_16x16x4_f32(false, a, false, bfr,
                                                        (short)0, acc, false, false);
        }
        float bb = (lm < NBASIS) ? sB3[lm] : 0.0f;
        #pragma unroll
        for (int rr = 0; rr < 8; ++rr)
            sC[(mrow + rr) * 16 + lm] = acc[rr] + bb;
    }
    __syncthreads();

    if (tid < RPB) { // softmax over 15 mixture logits per row
        float mx = -1e30f;
        for (int k = 0; k < NBASIS; ++k) mx = fmaxf(mx, sC[tid * 16 + k]);
        float ex[NBASIS];
        float s = 0.0f;
        for (int k = 0; k < NBASIS; ++k) { ex[k] = expf(sC[tid * 16 + k] - mx); s += ex[k]; }
        float inv = 1.0f / s;
        for (int k = 0; k < NBASIS; ++k) sC[tid * 16 + k] = ex[k] * inv;
        sC[tid * 16 + 15] = 0.0f;
    }

    // drain the async basis staging (per-wave count), then barrier so every
    // wave sees all LDS writes before the expansion reads them
#ifdef HAVE_ASYNC_LDS
#if defined(__has_builtin) && __has_builtin(__builtin_amdgcn_s_wait_asynccnt)
    __builtin_amdgcn_s_wait_asynccnt(0);
#else
    asm volatile("s_wait_asynccnt 0x0" ::: "memory");
#endif
#endif
    __syncthreads();

    // ---- phases 3/4: band expansion + banded products, chunked by 16 rows ----
    for (int ic = 0; ic < 4; ++ic) {
        int e0 = rs[ic * CHUNK_I];
        int ec = rs[ic * CHUNK_I + CHUNK_I] - e0;  // <= 112

        for (int w = tid; w < 4 * ec; w += THREADS) {
            int X  = w / ec;
            int ee = w - X * ec;
            int e  = e0 + ee;
#ifdef HAVE_ASYNC_LDS
            const float* bp = sBAS + X * (NBASIS * NE);
#else
            const float* bp = (X == 0) ? tm11b : (X == 1) ? tm12b
                             : (X == 2) ? tm21b : tm22b;
#endif
            float bv[NBASIS];
            #pragma unroll
            for (int k = 0; k < NBASIS; ++k) bv[k] = bp[k * NE + e];
            float dadd = (eDiag[e] && (X == 0 || X == 3)) ? 1.0f : 0.0f;
            for (int r = 0; r < RPB; ++r) {
                float s = dadd;
                #pragma unroll
                for (int k = 0; k < NBASIS; ++k) s += sC[r * 16 + k] * bv[k];
                sTM[(X * RPB + r) * CEMAX + ee] = s;
            }
        }
        __syncthreads();

        for (int w = tid; w < RPB * CHUNK_I; w += THREADS) {
            int r  = w >> 4;
            int ii = w & 15;
            int i  = ic * CHUNK_I + ii;
            int jl = (i - BWD < 0) ? 0 : i - BWD;
            int jh = (i + BWD > LOD - 1) ? LOD - 1 : i + BWD;
            int ebase = rs[i] - e0;
            float nmu = 0.f, nml = 0.f, ncu = 0.f, ncl = 0.f, ncs = 0.f;
            for (int j = jl; j <= jh; ++j) {
                int ee = ebase + (j - jl);
                float t11 = sTM[(0 * RPB + r) * CEMAX + ee];
                float t12 = sTM[(1 * RPB + r) * CEMAX + ee];
                float t21 = sTM[(2 * RPB + r) * CEMAX + ee];
                float t22 = sTM[(3 * RPB + r) * CEMAX + ee];
                float mu = sPM[r * LSD + j];
                float ml = sPM[r * LSD + LOD + j];
                float cu = sPCU[r * LOD + j];
                float cs = sPCS[r * LOD + j];
                float cl = sPCL[r * LOD + j];
                nmu += t11 * mu + t12 * ml;
                nml += t21 * mu + t22 * ml;
                ncu += t11 * t11 * cu + 2.0f * t11 * t12 * cs + t12 * t12 * cl;
                ncl += t21 * t21 * cu + 2.0f * t21 * t22 * cs + t22 * t22 * cl;
                ncs += t21 * t11 * cu + (t22 * t11 + t21 * t12) * cs + t22 * t12 * cl;
            }
            float tcu = log1pf(expf(ltn[i]));
            float tcl = log1pf(expf(ltn[LOD + i]));
            int row = r0 + r;
            out[OFF_NM + row * LSD + i] = nmu;
            out[OFF_NM + row * LSD + LOD + i] = nml;
            out[OFF_NCU + row * LOD + i] = ncu + tcu;
            out[OFF_NCL + row * LOD + i] = ncl + tcl;
            out[OFF_NCS + row * LOD + i] = ncs;
        }
        __syncthreads();
    }
}

extern "C" void kernel_launch(void* const* d_in, const int* in_sizes, int n_in,
                              void* d_out, int out_size, void* d_ws, size_t ws_size,
                              hipStream_t stream) {
    (void)in_sizes; (void)n_in; (void)out_size; (void)d_ws; (void)ws_size;
    const float* prior_mean = (const float*)d_in[0];
    const float* cov_u      = (const float*)d_in[1];
    const float* cov_l      = (const float*)d_in[2];
    const float* cov_s      = (const float*)d_in[3];
    const float* obs        = (const float*)d_in[4];
    const float* obs_var    = (const float*)d_in[5];
    const float* W1         = (const float*)d_in[6];
    const float* b1         = (const float*)d_in[7];
    const float* W2         = (const float*)d_in[8];
    const float* b2         = (const float*)d_in[9];
    const float* W3         = (const float*)d_in[10];
    const float* b3         = (const float*)d_in[11];
    const float* tm11b      = (const float*)d_in[12];
    const float* tm12b      = (const float*)d_in[13];
    const float* tm21b      = (const float*)d_in[14];
    const float* tm22b      = (const float*)d_in[15];
    const float* ltn        = (const float*)d_in[16];
    float* out = (float*)d_out;

    size_t shmem = (size_t)LDS_FLOATS * sizeof(float);   // ~210 KB < 320 KB/WGP
    dim3 grid(BATCH / RPB);   // 256 blocks of 128 threads (4 wave32s)
    dim3 block(THREADS);
    rkn_cell_fused<<<grid, block, shmem, stream>>>(
        prior_mean, cov_u, cov_l, cov_s, obs, obs_var,
        W1, b1, W2, b2, W3, b3,
        tm11b, tm12b, tm21b, tm22b, ltn, out);
}